// Net_29154238005503
// MI455X (gfx1250) — compile-verified
//
#include <hip/hip_runtime.h>

// SLIDE-style 2-layer sparse net for MI455X (gfx1250, wave32, WMMA).
// Layer2 moves ~268MB of gathered W2 rows -> memory bound (~11.5us @ 23.3TB/s).
// Use V_WMMA_F32_16X16X4_F32 (full f32 precision, matrix pipe does reductions).

typedef __attribute__((ext_vector_type(2))) float v2f;
typedef __attribute__((ext_vector_type(8))) float v8f;

#define B_DIM   128
#define NNZ     128
#define F_DIM_  135909
#define H_DIM   128
#define C_DIM   670091
#define KOUT    4096

// Extract accumulator element for this lane's output row.
// D layout (16x16 f32): lanes 0-15 hold M=0..7 in vgprs0..7 (N=lane),
// lanes 16-31 hold M=8..15. B was replicated across N, so any N is valid.
__device__ __forceinline__ float pick_acc(v8f acc, int s) {
    float r = (s == 0) ? acc[0]
            : (s == 1) ? acc[1]
            : (s == 2) ? acc[2]
            : (s == 3) ? acc[3]
            : (s == 4) ? acc[4]
            : (s == 5) ? acc[5]
            : (s == 6) ? acc[6]
            :            acc[7];
    return r;
}

// ---------------------------------------------------------------------------
// Layer 1: val1[b,h] = relu( sum_n inv[b,n] * W1[h, idx[b,n]] + b1[h] )
// One wave per (b, 16-wide h tile). A = gathered W1 elements, B = inv chunk
// replicated across N. 32 chained wmma_f32_16x16x4_f32.
// ---------------------------------------------------------------------------
__global__ void __launch_bounds__(256)
slide_layer1(const float* __restrict__ inv,
             const int*   __restrict__ iidx,
             const float* __restrict__ W1,
             const float* __restrict__ b1,
             float*       __restrict__ val1) {
    const int lane  = threadIdx.x & 31;
    const int wave  = threadIdx.x >> 5;
    const int tile  = blockIdx.x * 8 + wave;       // 8 waves / block
    const int b     = tile >> 3;                   // H/16 = 8 tiles per batch
    const int ht    = tile & 7;

    const int m     = lane & 15;                   // A-matrix row (M)
    const int khalf = lane >> 4;                   // K sub-pair: 0 -> K0,K1 ; 1 -> K2,K3
    const int h     = ht * 16 + m;

    const float* wrow = W1  + (size_t)h * F_DIM_;  // row h of W1 (we gather columns)
    const float* vrow = inv + b * NNZ;
    const int*   irow = iidx + b * NNZ;

    v8f acc = {};
#pragma unroll
    for (int kc = 0; kc < NNZ; kc += 4) {
        const int c0 = irow[kc + 2 * khalf + 0];
        const int c1 = irow[kc + 2 * khalf + 1];
        v2f a;
        a[0] = wrow[c0];                           // A[m][kc+2*khalf+0]
        a[1] = wrow[c1];                           // A[m][kc+2*khalf+1]
        v2f bb = *(const v2f*)(vrow + kc + 2 * khalf);  // B chunk (replicated over N)
        acc = __builtin_amdgcn_wmma_f32_16x16x4_f32(
            /*neg_a=*/false, a, /*neg_b=*/false, bb,
            /*c_mod=*/(short)0, acc, /*reuse_a=*/false, /*reuse_b=*/false);
    }

    // Writers: lanes 0-7 (M = lane), lanes 16-23 (M = 8 + (lane&7)).
    if ((lane & 8) == 0) {
        const int s     = lane & 7;
        const int m_out = ((lane & 16) >> 1) | s;  // 0..7 or 8..15
        const int h_out = ht * 16 + m_out;
        const float r   = pick_acc(acc, s);
        val1[b * H_DIM + h_out] = fmaxf(r + b1[h_out], 0.0f);
    }
}

// ---------------------------------------------------------------------------
// Layer 2: val2[b,k] = <val1[b,:], W2[lbl[b,k],:]> + b2[lbl[b,k]]
// One wave per (b, 16-wide k tile). A = 16 gathered W2 rows (contiguous 512B
// rows -> streaming gather), B = val1[b] chunk replicated across N.
// ---------------------------------------------------------------------------
__global__ void __launch_bounds__(256)
slide_layer2(const float* __restrict__ val1,
             const int*   __restrict__ lbl,
             const float* __restrict__ W2,
             const float* __restrict__ b2,
             float*       __restrict__ out) {
    const int lane  = threadIdx.x & 31;
    const int wave  = threadIdx.x >> 5;
    const int tile  = blockIdx.x * 8 + wave;       // 8 waves / block
    const int b     = tile >> 8;                   // KOUT/16 = 256 tiles per batch
    const int kt    = tile & 255;

    const int m     = lane & 15;
    const int khalf = lane >> 4;

    const int row = lbl[b * KOUT + kt * 16 + m];
    const float* rowp = W2 + (size_t)row * H_DIM;  // contiguous 512B row
    const float* vrow = val1 + b * H_DIM;
    __builtin_prefetch(rowp, 0, 0);                // global_prefetch_b8

    v8f acc = {};
#pragma unroll
    for (int kc = 0; kc < H_DIM; kc += 4) {
        v2f a  = *(const v2f*)(rowp + kc + 2 * khalf);  // A[m][k..k+1], 8B aligned
        v2f bb = *(const v2f*)(vrow + kc + 2 * khalf);  // B replicated over N
        acc = __builtin_amdgcn_wmma_f32_16x16x4_f32(
            false, a, false, bb, (short)0, acc, false, false);
    }

    if ((lane & 8) == 0) {
        const int s     = lane & 7;
        const int m_out = ((lane & 16) >> 1) | s;
        const int gk    = b * KOUT + kt * 16 + m_out;
        const int row_o = lbl[gk];
        out[gk] = pick_acc(acc, s) + b2[row_o];
    }
}

// Second tuple output: bit-copy int32 label indices after the B*KOUT floats.
__global__ void __launch_bounds__(256)
copy_idx(const int* __restrict__ lbl, int* __restrict__ out_i) {
    const int i = blockIdx.x * 256 + threadIdx.x;
    out_i[i] = lbl[i];
}

extern "C" void kernel_launch(void* const* d_in, const int* in_sizes, int n_in,
                              void* d_out, int out_size, void* d_ws, size_t ws_size,
                              hipStream_t stream) {
    const float* in_values = (const float*)d_in[0];   // [B, NNZ] f32
    const int*   in_idx    = (const int*)  d_in[1];   // [B, NNZ] i32
    const int*   lbl       = (const int*)  d_in[2];   // [B, KOUT] i32
    const float* W1        = (const float*)d_in[3];   // [H, F_DIM] f32
    const float* b1        = (const float*)d_in[4];   // [H] f32
    const float* W2        = (const float*)d_in[5];   // [C, H] f32
    const float* b2        = (const float*)d_in[6];   // [C] f32

    float* out  = (float*)d_out;
    float* val1 = (float*)d_ws;                       // [B, H] = 64 KB scratch

    // Layer 1: B*(H/16) = 1024 waves -> 128 blocks of 8 waves.
    slide_layer1<<<(B_DIM * (H_DIM / 16)) / 8, 256, 0, stream>>>(
        in_values, in_idx, W1, b1, val1);

    // Layer 2: B*(KOUT/16) = 32768 waves -> 4096 blocks of 8 waves.
    slide_layer2<<<(B_DIM * (KOUT / 16)) / 8, 256, 0, stream>>>(
        val1, lbl, W2, b2, out);

    // Tuple output #2: the label indices themselves.
    copy_idx<<<(B_DIM * KOUT) / 256, 256, 0, stream>>>(
        lbl, (int*)(out + (size_t)B_DIM * KOUT));
}